// DBSCAN_70540542869835
// MI455X (gfx1250) — compile-verified
//
#include <hip/hip_runtime.h>
#include <hip/hip_bf16.h>

typedef __attribute__((ext_vector_type(2))) float v2f;
typedef __attribute__((ext_vector_type(8))) float v8f;

#define NPTS   10000
#define DIMF   64
#define NTILE  625          // NPTS / 16
#define NPAIR  313          // ceil(NTILE / 2)  -> one 32-bit bitmask word per pair
#define WPR    320          // bitmask words per row (padded stride; only 0..312 written/read)
#define EPSV   0.4f
#define PROP_ITERS 24

// ---------------------------------------------------------------------------
// 1) Row-normalize: one wave per row (64 floats -> 2 per lane).
// ---------------------------------------------------------------------------
__global__ void normalize_rows(const float* __restrict__ x, float* __restrict__ nf) {
    int wave = threadIdx.x >> 5;
    int lane = threadIdx.x & 31;
    int row  = blockIdx.x * 8 + wave;
    if (row >= NPTS) return;
    const float* xr = x + (size_t)row * DIMF;
    float a = xr[lane];
    float b = xr[lane + 32];
    float s = a * a + b * b;
    #pragma unroll
    for (int off = 16; off; off >>= 1) s += __shfl_xor(s, off, 32);
    float inv = 1.0f / sqrtf(s);
    float* o = nf + (size_t)row * DIMF;
    o[lane]      = a * inv;
    o[lane + 32] = b * inv;
}

// ---------------------------------------------------------------------------
// 2) Gram matrix tile (fp32 WMMA) + threshold -> bitmask adjacency.
//    One wave computes a 16x32 tile (tile-pair) so it owns whole 32-bit words.
//    For the last (odd) pair, brow1 aliases brow0; the bogus acc1 result is
//    discarded at the ballot stage, keeping the k-loop branch-free.
// ---------------------------------------------------------------------------
__global__ void gram_bits(const float* __restrict__ nf, unsigned* __restrict__ bits) {
    int wave = threadIdx.x >> 5;
    int lane = threadIdx.x & 31;
    long tid = (long)blockIdx.x * 8 + wave;            // tile-pair id
    const long NPAIRS_TOTAL = (long)NTILE * NPAIR;     // 625 * 313
    if (tid >= NPAIRS_TOTAL) return;                   // wave-uniform exit (EXEC stays all-1s)

    int ti  = (int)(tid / NPAIR);      // row-tile
    int tp  = (int)(tid % NPAIR);      // word index within row == tj0/2
    int tj0 = tp * 2;
    int tj1 = tj0 + 1;
    const bool has1 = (tj1 < NTILE);   // last pair covers only one column tile

    int m     = lane & 15;
    int khalf = (lane >> 4) << 1;      // lanes 0-15 -> K={0,1}; lanes 16-31 -> K={2,3}

    const float* arow  = nf + (size_t)(ti  * 16 + m) * DIMF;
    const float* brow0 = nf + (size_t)(tj0 * 16 + m) * DIMF;
    const float* brow1 = has1 ? (nf + (size_t)(tj1 * 16 + m) * DIMF) : brow0;

    v8f acc0 = {};
    v8f acc1 = {};
    #pragma unroll
    for (int kc = 0; kc < DIMF / 4; ++kc) {            // 16 K-steps of 4
        int k = kc * 4 + khalf;
        v2f a  = *(const v2f*)(arow  + k);
        v2f b0 = *(const v2f*)(brow0 + k);
        v2f b1 = *(const v2f*)(brow1 + k);             // unconditional: no exec-mask branches
        acc0 = __builtin_amdgcn_wmma_f32_16x16x4_f32(false, a, false, b0,
                                                     (short)0, acc0, false, false);
        acc1 = __builtin_amdgcn_wmma_f32_16x16x4_f32(false, a, false, b1,
                                                     (short)0, acc1, false, false);
    }

    // C/D layout: VGPR r, lanes 0-15 -> (M=r, N=lane); lanes 16-31 -> (M=r+8, N=lane-16).
    // ballot low16  = row (ti*16+r)   bits; ballot high16 = row (ti*16+8+r) bits.
    unsigned myword = 0;
    #pragma unroll
    for (int r = 0; r < 8; ++r) {
        unsigned p0 = (unsigned)__ballot(acc0[r] > EPSV);
        unsigned p1 = has1 ? (unsigned)__ballot(acc1[r] > EPSV) : 0u;
        unsigned wlo = (p0 & 0xFFFFu) | ((p1 & 0xFFFFu) << 16);   // row r
        unsigned whi = (p0 >> 16)     | (p1 & 0xFFFF0000u);       // row r+8
        if (lane == r)     myword = wlo;
        if (lane == r + 8) myword = whi;
    }
    if (lane < 16)
        bits[(size_t)(ti * 16 + lane) * WPR + tp] = myword;
}

// ---------------------------------------------------------------------------
// 3) Connected components: min-label Jacobi propagation + pointer jumping.
// ---------------------------------------------------------------------------
__global__ void init_comp(int* __restrict__ comp) {
    int i = blockIdx.x * 256 + threadIdx.x;
    if (i < NPTS) comp[i] = i;
}

__global__ void propagate(const unsigned* __restrict__ bits,
                          const int* __restrict__ cin,
                          int* __restrict__ cout) {
    int wave = threadIdx.x >> 5;
    int lane = threadIdx.x & 31;
    int row  = blockIdx.x * 8 + wave;
    if (row >= NPTS) return;
    const unsigned* wrow = bits + (size_t)row * WPR;
    int mv = cin[row];
    for (int w = lane; w < NPAIR; w += 32) {           // only words 0..312 are valid
        unsigned word = wrow[w];
        while (word) {
            int b = __ffs(word) - 1;
            word &= word - 1;
            int j = (w << 5) + b;
            mv = min(mv, cin[j]);
        }
    }
    #pragma unroll
    for (int off = 16; off; off >>= 1) mv = min(mv, __shfl_xor(mv, off, 32));
    mv = min(mv, cin[mv]);                             // pointer-jump shortcuts
    mv = min(mv, cin[mv]);
    if (lane == 0) cout[row] = mv;
}

// ---------------------------------------------------------------------------
// 4) Rank roots (exclusive count of comp[i]==i) via single-block scan.
// ---------------------------------------------------------------------------
__global__ void scan_ranks(const int* __restrict__ comp, int* __restrict__ ranks) {
    __shared__ int sdata[1024];
    __shared__ int soffset;
    if (threadIdx.x == 0) soffset = 0;
    __syncthreads();
    for (int base = 0; base < NPTS; base += 1024) {
        int i = base + (int)threadIdx.x;
        int v = (i < NPTS && comp[i] == i) ? 1 : 0;
        sdata[threadIdx.x] = v;
        __syncthreads();
        for (int off = 1; off < 1024; off <<= 1) {
            int t = (threadIdx.x >= (unsigned)off) ? sdata[threadIdx.x - off] : 0;
            __syncthreads();
            sdata[threadIdx.x] += t;
            __syncthreads();
        }
        if (i < NPTS) ranks[i] = soffset + sdata[threadIdx.x] - 1;
        __syncthreads();
        if (threadIdx.x == 0) soffset += sdata[1023];
        __syncthreads();
    }
}

__global__ void assign_labels(const int* __restrict__ comp,
                              const int* __restrict__ ranks,
                              int* __restrict__ labels) {
    int i = blockIdx.x * 256 + threadIdx.x;
    if (i < NPTS) labels[i] = ranks[comp[i]];
}

// ---------------------------------------------------------------------------
// Launch
// ---------------------------------------------------------------------------
extern "C" void kernel_launch(void* const* d_in, const int* in_sizes, int n_in,
                              void* d_out, int out_size, void* d_ws, size_t ws_size,
                              hipStream_t stream) {
    const float* x  = (const float*)d_in[0];
    int* labels     = (int*)d_out;                  // reference output dtype: int32

    // workspace layout:
    //   nf:    10000*64 f32            =  2.56 MB
    //   bits:  10000*320 u32 (bitmask) = 12.80 MB (L2-resident)
    //   compA/compB/ranks: 3 * 10016 i32
    char* ws = (char*)d_ws;
    float*    nf    = (float*)ws;
    size_t    nfB   = (size_t)NPTS * DIMF * sizeof(float);
    unsigned* bits  = (unsigned*)(ws + nfB);
    size_t    bitsB = (size_t)NPTS * WPR * sizeof(unsigned);
    int*      compA = (int*)(ws + nfB + bitsB);
    int*      compB = compA + 10016;
    int*      ranks = compB + 10016;

    normalize_rows<<<(NPTS + 7) / 8, 256, 0, stream>>>(x, nf);

    long pairs = (long)NTILE * NPAIR;               // 195625 wave-tiles
    int  g2    = (int)((pairs + 7) / 8);
    gram_bits<<<g2, 256, 0, stream>>>(nf, bits);

    init_comp<<<(NPTS + 255) / 256, 256, 0, stream>>>(compA);
    for (int it = 0; it < PROP_ITERS; ++it) {
        const int* cin = (it & 1) ? compB : compA;
        int*       cot = (it & 1) ? compA : compB;
        propagate<<<(NPTS + 7) / 8, 256, 0, stream>>>(bits, cin, cot);
    }
    // PROP_ITERS is even -> final labels in compA

    scan_ranks<<<1, 1024, 0, stream>>>(compA, ranks);
    assign_labels<<<(NPTS + 255) / 256, 256, 0, stream>>>(compA, ranks, labels);
}